// GCN_32676111188646
// MI455X (gfx1250) — compile-verified
//
#include <hip/hip_runtime.h>
#include <hip/hip_bf16.h>

// ---------------- problem constants ----------------
#define NN     50000     // nodes
#define NE     800000    // edges
#define NG     2048      // graphs
#define INDIM  128
#define HID    256
#define NLAY   4
#define BN_EPS 1e-5f

typedef float v2f __attribute__((ext_vector_type(2)));
typedef float v4f __attribute__((ext_vector_type(4)));
typedef float v8f __attribute__((ext_vector_type(8)));

// Hardware no-return f32 atomic add (CDNA5 global_atomic_add_f32, STOREcnt-tracked;
// completion guaranteed by implicit wait-idle at s_endpgm).
__device__ __forceinline__ void atomAddF(float* p, float v) {
    asm volatile("global_atomic_add_f32 %0, %1, off" :: "v"(p), "v"(v) : "memory");
}

// ---------------- utility kernels ----------------
__global__ void zero_f(float* p, int n) {
    int i = blockIdx.x * blockDim.x + threadIdx.x;
    if (i < n) p[i] = 0.0f;
}
__global__ void zero_i(int* p, int n) {
    int i = blockIdx.x * blockDim.x + threadIdx.x;
    if (i < n) p[i] = 0;
}

// degree over dst (+1 self loop applied in finalize)
__global__ void deg_count(const int* __restrict__ dst, int* __restrict__ degi, int e_cnt) {
    int i = blockIdx.x * blockDim.x + threadIdx.x;
    if (i < e_cnt) atomicAdd(&degi[dst[i]], 1);
}
__global__ void deg_finalize(const int* __restrict__ degi, float* __restrict__ dinv,
                             float* __restrict__ selfn, int n) {
    int i = blockIdx.x * blockDim.x + threadIdx.x;
    if (i < n) {
        float d = (float)(degi[i] + 1);     // +1 self loop
        dinv[i]  = rsqrtf(d);
        selfn[i] = 1.0f / d;
    }
}

// ---------------- WMMA FP32 GEMM: Hout[M,Nc] = X[M,K] @ W[K,Nc] ----------------
// block = 256 threads = 8 waves. blockIdx.y selects a 16-col slab of W (staged
// in LDS); each wave owns one 16-row M tile -> one v_wmma_f32_16x16x4_f32 per
// 4-wide K step.
__global__ void wmma_gemm_f32(const float* __restrict__ X, const float* __restrict__ W,
                              float* __restrict__ Hout, int M, int K, int Nc) {
    __shared__ float lds[HID * 16];          // up to 256x16 slab of W (16 KB)
    const int n0 = blockIdx.y * 16;
    for (int t = threadIdx.x; t < K * 16; t += blockDim.x) {
        int kk = t >> 4, j = t & 15;
        lds[t] = W[(size_t)kk * Nc + n0 + j];
    }
    __syncthreads();

    const int wave  = threadIdx.x >> 5;
    const int lane  = threadIdx.x & 31;
    const int mtile = blockIdx.x * 8 + wave;
    if (mtile * 16 >= M) return;             // whole-wave uniform exit

    const int m  = mtile * 16 + (lane & 15); // A row this lane feeds
    const int kb = (lane >> 4) << 1;         // K pair: lanes 0-15 -> {0,1}, 16-31 -> {2,3}
    const int n  = lane & 15;                // B/C column this lane feeds

    const float* xr = X + (size_t)m * K;
    v8f acc = {0.f, 0.f, 0.f, 0.f, 0.f, 0.f, 0.f, 0.f};

    #pragma unroll 4
    for (int k = 0; k < K; k += 4) {
        v2f a = *(const v2f*)(xr + k + kb);                     // A[m][k+kb..k+kb+1]
        v2f b = { lds[(k + kb) * 16 + n],                       // W[k+kb  ][n0+n]
                  lds[(k + kb + 1) * 16 + n] };                 // W[k+kb+1][n0+n]
        acc = __builtin_amdgcn_wmma_f32_16x16x4_f32(
                  false, a, false, b, (short)0, acc, false, false);
    }

    // C/D layout: acc[i] = D[M = i + 8*(lane>>4)][N = lane&15]
    const int rbase = mtile * 16 + ((lane >> 4) << 3);
    float* orow = Hout + (size_t)rbase * Nc + n0 + n;
    #pragma unroll
    for (int i = 0; i < 8; ++i) orow[(size_t)i * Nc] = acc[i];
}

// agg = h * self_norm[node] + bias[feat], 16B vectorized (fully overwrites agg)
__global__ void init_agg_v4(const float* __restrict__ h, const float* __restrict__ selfn,
                            const float* __restrict__ bias, float* __restrict__ agg, int total4) {
    int i = blockIdx.x * blockDim.x + threadIdx.x;
    if (i < total4) {
        int nidx = i >> 6;            // (i*4) / HID
        int f    = (i << 2) & 255;    // (i*4) % HID
        v4f hv = *(const v4f*)(h + ((size_t)i << 2));
        v4f bv = *(const v4f*)(bias + f);
        v4f r  = hv * selfn[nidx] + bv;
        *(v4f*)(agg + ((size_t)i << 2)) = r;
    }
}

// one wave per edge: agg[dst] += h[src] * dinv[src]*dinv[dst]
__global__ void edge_scatter(const int* __restrict__ src, const int* __restrict__ dst,
                             const float* __restrict__ dinv, const float* __restrict__ h,
                             float* __restrict__ agg, int e_cnt) {
    int e = blockIdx.x * 8 + (threadIdx.x >> 5);
    if (e >= e_cnt) return;
    int lane = threadIdx.x & 31;
    int s = src[e], d = dst[e];
    float w = dinv[s] * dinv[d];
    const float* hs = h + (size_t)s * HID;
    float* ad = agg + (size_t)d * HID;
    #pragma unroll
    for (int j = 0; j < 8; ++j) {
        int f = j * 32 + lane;                 // coalesced 128B gather per step
        atomAddF(ad + f, hs[f] * w);
    }
}

// per-layer BN constants: scale = gamma*rsqrt(var+eps), shift = beta - mean*scale
__global__ void bn_prep(const float* __restrict__ gamma, const float* __restrict__ beta,
                        const float* __restrict__ mean, const float* __restrict__ var,
                        float* __restrict__ scale, float* __restrict__ shift) {
    int i = threadIdx.x;   // HID threads
    float s = gamma[i] * rsqrtf(var[i] + BN_EPS);
    scale[i] = s;
    shift[i] = beta[i] - mean[i] * s;
}

// eval-mode BN + ReLU, in place, 16B vectorized
__global__ void bn_relu_v4(float* __restrict__ x, const float* __restrict__ scale,
                           const float* __restrict__ shift, int total4) {
    int i = blockIdx.x * blockDim.x + threadIdx.x;
    if (i < total4) {
        int f = (i << 2) & 255;
        v4f xv = *(const v4f*)(x + ((size_t)i << 2));
        v4f sc = *(const v4f*)(scale + f);
        v4f sh = *(const v4f*)(shift + f);
        v4f v  = xv * sc + sh;
        #pragma unroll
        for (int c = 0; c < 4; ++c) v[c] = v[c] > 0.f ? v[c] : 0.f;
        *(v4f*)(x + ((size_t)i << 2)) = v;
    }
}

// ---------------- pooling ----------------
__global__ void count_nodes(const int* __restrict__ batch, float* __restrict__ cnt, int n) {
    int i = blockIdx.x * blockDim.x + threadIdx.x;
    if (i < n) atomAddF(&cnt[batch[i]], 1.0f);
}
__global__ void pool_sum(const float* __restrict__ x, const int* __restrict__ batch,
                         float* __restrict__ pooled, int n) {
    int nd = blockIdx.x * 8 + (threadIdx.x >> 5);
    if (nd >= n) return;
    int lane = threadIdx.x & 31;
    int g = batch[nd];
    const float* xr = x + (size_t)nd * HID;
    float* pg = pooled + (size_t)g * HID;
    #pragma unroll
    for (int j = 0; j < 8; ++j) {
        int f = j * 32 + lane;
        atomAddF(pg + f, xr[f]);
    }
}
__global__ void pool_div_v4(float* __restrict__ pooled, const float* __restrict__ cnt, int total4) {
    int i = blockIdx.x * blockDim.x + threadIdx.x;
    if (i < total4) {
        float inv = 1.0f / fmaxf(cnt[i >> 6], 1.0f);
        v4f p = *(const v4f*)(pooled + ((size_t)i << 2));
        *(v4f*)(pooled + ((size_t)i << 2)) = p * inv;
    }
}

// ---------------- final MLP tail: out[g] = relu(h1+b1) @ lin2W + b2 ----------------
__global__ void final_mlp(const float* __restrict__ h1, const float* __restrict__ lin1b,
                          const float* __restrict__ lin2W, const float* __restrict__ lin2b,
                          float* __restrict__ out, int g_cnt) {
    int g = blockIdx.x * blockDim.x + threadIdx.x;
    if (g >= g_cnt) return;
    const float* r = h1 + (size_t)g * (HID / 2);
    float s = lin2b[0];
    #pragma unroll 4
    for (int f = 0; f < HID / 2; ++f) {
        float v = r[f] + lin1b[f];
        v = v > 0.f ? v : 0.f;
        s += v * lin2W[f];
    }
    out[g] = s;
}

// ---------------- host launch ----------------
extern "C" void kernel_launch(void* const* d_in, const int* in_sizes, int n_in,
                              void* d_out, int out_size, void* d_ws, size_t ws_size,
                              hipStream_t stream) {
    const float* x_in   = (const float*)d_in[0];
    const int*   eidx   = (const int*)  d_in[1];   // [2, E]: row0=src, row1=dst
    const int*   batch  = (const int*)  d_in[2];
    const float* convW0 = (const float*)d_in[3];
    const float* convWs = (const float*)d_in[4];   // [3, 256, 256]
    const float* convB  = (const float*)d_in[5];   // [4, 256]
    const float* gamma  = (const float*)d_in[6];
    const float* beta   = (const float*)d_in[7];
    const float* mean   = (const float*)d_in[8];
    const float* var    = (const float*)d_in[9];
    const float* lin1W  = (const float*)d_in[10];  // [256, 128]
    const float* lin1b  = (const float*)d_in[11];
    const float* lin2W  = (const float*)d_in[12];  // [128, 1]
    const float* lin2b  = (const float*)d_in[13];
    float* out = (float*)d_out;

    const int* src = eidx;
    const int* dst = eidx + NE;

    // workspace layout (ping-pong big buffers + small scratch)
    float* ws = (float*)d_ws;
    size_t off = 0;
    float* bufA   = ws + off; off += (size_t)NN * HID;   // agg / x
    float* bufB   = ws + off; off += (size_t)NN * HID;   // gemm output h
    float* dinv   = ws + off; off += NN;
    float* selfn  = ws + off; off += NN;
    int*   degi   = (int*)(ws + off); off += NN;
    float* pooled = ws + off; off += (size_t)NG * HID;
    float* cnt    = ws + off; off += NG;                 // contiguous after pooled
    float* h1     = ws + off; off += (size_t)NG * (HID / 2);
    float* bnscale= ws + off; off += HID;
    float* bnshift= ws + off; off += HID;

    const int TOTAL  = NN * HID;         // 12.8M
    const int TOTAL4 = TOTAL / 4;        // 3.2M float4s
    const int EWB    = (NE + 7) / 8;     // edge-scatter blocks (wave per edge)

    // ---- degrees ----
    zero_i<<<(NN + 255) / 256, 256, 0, stream>>>(degi, NN);
    deg_count<<<(NE + 255) / 256, 256, 0, stream>>>(dst, degi, NE);
    deg_finalize<<<(NN + 255) / 256, 256, 0, stream>>>(degi, dinv, selfn, NN);

    // ---- GCN layers ----
    const float* Wl[NLAY] = { convW0, convWs, convWs + HID * HID, convWs + 2 * HID * HID };
    const int    Kl[NLAY] = { INDIM, HID, HID, HID };
    for (int l = 0; l < NLAY; ++l) {
        const float* xin = (l == 0) ? x_in : bufA;
        const int K = Kl[l];
        dim3 grid((NN / 16 + 7) / 8, HID / 16);              // (391, 16)
        wmma_gemm_f32<<<grid, 256, 0, stream>>>(xin, Wl[l], bufB, NN, K, HID);
        init_agg_v4<<<(TOTAL4 + 255) / 256, 256, 0, stream>>>(bufB, selfn, convB + l * HID,
                                                              bufA, TOTAL4);
        edge_scatter<<<EWB, 256, 0, stream>>>(src, dst, dinv, bufB, bufA, NE);
        bn_prep<<<1, HID, 0, stream>>>(gamma + l * HID, beta + l * HID,
                                       mean + l * HID, var + l * HID, bnscale, bnshift);
        bn_relu_v4<<<(TOTAL4 + 255) / 256, 256, 0, stream>>>(bufA, bnscale, bnshift, TOTAL4);
    }

    // ---- global mean pool ----
    zero_f<<<(NG * HID + NG + 255) / 256, 256, 0, stream>>>(pooled, NG * HID + NG); // pooled+cnt
    count_nodes<<<(NN + 255) / 256, 256, 0, stream>>>(batch, cnt, NN);
    pool_sum<<<(NN + 7) / 8, 256, 0, stream>>>(bufA, batch, pooled, NN);
    pool_div_v4<<<(NG * HID / 4 + 255) / 256, 256, 0, stream>>>(pooled, cnt, NG * HID / 4);

    // ---- MLP head: h1 = pooled @ lin1W  (WMMA), then relu+dot in tail ----
    dim3 grid1((NG / 16 + 7) / 8, (HID / 2) / 16);           // (16, 8)
    wmma_gemm_f32<<<grid1, 256, 0, stream>>>(pooled, lin1W, h1, NG, HID, HID / 2);
    final_mlp<<<(NG + 255) / 256, 256, 0, stream>>>(h1, lin1b, lin2W, lin2b, out, NG);
}